// BasisVQ_63780264346098
// MI455X (gfx1250) — compile-verified
//
#include <hip/hip_runtime.h>
#include <hip/hip_bf16.h>
#include <stdint.h>

#define C_DIM 1024          // codes per row
#define D_DIM 900           // basis atom dim (225 float4, 16B aligned rows)
#define NT 256
#define NWAVES (NT / 32)
#define ROWS_PER_WAVE 8
#define TILE_ROWS 2
#define TILES (ROWS_PER_WAVE / TILE_ROWS)   // 4 tiles of 2 rows

typedef float v4f  __attribute__((ext_vector_type(4)));
typedef unsigned int u32x4 __attribute__((ext_vector_type(4)));
typedef unsigned int u32x8 __attribute__((ext_vector_type(8)));

// ---- CDNA5 Tensor Data Mover: stage a 2-row (2 x 1024 f32 = 8KB) tile ------
// D# group0 (4 SGPRs) + group1 (8 SGPRs); groups 2/3 omitted (2-D tensor).
// TDM walks X (1024 elems) then Y (2 rows), writing LDS linearly from lds_addr.
__device__ __forceinline__ void tdm_load_tile(unsigned lds_addr, const float* gaddr) {
    const unsigned lo = (unsigned)(uintptr_t)gaddr;
    const unsigned hi = (unsigned)((uintptr_t)gaddr >> 32);

    u32x4 g0;
    g0[0] = 1u;                                    // count=1 (valid user D#)
    g0[1] = lds_addr;                              // LDS byte address
    g0[2] = lo;                                    // global_addr[31:0]
    g0[3] = (hi & 0x01FFFFFFu) | 0x80000000u;      // global_addr[56:32] | type=2

    u32x8 g1;
    g1[0] = 0x00020000u;                           // mask=0, data_size=2 (4B)
    g1[1] = ((unsigned)C_DIM & 0xFFFFu) << 16;     // tensor_dim0[15:0] @ bits63:48
    g1[2] = ((unsigned)TILE_ROWS) << 16;           // dim0 hi=0 | tensor_dim1 lo16
    g1[3] = ((unsigned)C_DIM) << 16;               // dim1 hi=0 | tile_dim0=1024
    g1[4] = (unsigned)TILE_ROWS;                   // tile_dim1=2, tile_dim2=0
    g1[5] = (unsigned)C_DIM;                       // tensor_dim0_stride lo32
    g1[6] = 0u;                                    // stride hi, dim1_stride lo
    g1[7] = 0u;

    asm volatile("tensor_load_to_lds %0, %1"
                 :
                 : "s"(g0), "s"(g1)
                 : "memory");
}

template <int N>
__device__ __forceinline__ void wait_tensor() {
    asm volatile("s_wait_tensorcnt %0" :: "n"(N) : "memory");
}

__global__ __launch_bounds__(NT) void BasisVQ_tdm_rows(
    const float* __restrict__ latent,   // [BK, 1024]
    const float* __restrict__ basis,    // [1024, 900]
    float* __restrict__ out_q,          // [BK, 900]
    float* __restrict__ out_idx)        // [BK] (argmax, stored as float)
{
    // Per-wave double-buffered 2-row tiles: 8 waves x 2 x 8KB = 128KB / block.
    __shared__ __align__(16) float buf[NWAVES][2][TILE_ROWS * C_DIM];

    const int lane = threadIdx.x & 31;
    const int wave = threadIdx.x >> 5;
    const int gw   = blockIdx.x * NWAVES + wave;   // global wave id
    const int row0 = gw * ROWS_PER_WAVE;

    // Low 32 bits of a generic shared pointer == LDS byte offset.
    const unsigned ldsA = (unsigned)(uintptr_t)(&buf[wave][0][0]);
    const unsigned ldsB = (unsigned)(uintptr_t)(&buf[wave][1][0]);

    const float* wbase = latent + (size_t)row0 * C_DIM;

    // Prologue: DMA tile 0.
    tdm_load_tile(ldsA, wbase);

    for (int t = 0; t < TILES; ++t) {
        if (t + 1 < TILES) {
            // Kick next tile's DMA, then wait until only it remains in flight:
            // tensor ops retire in order per wave => cnt<=1 means tile t landed.
            tdm_load_tile(((t + 1) & 1) ? ldsB : ldsA,
                          wbase + (size_t)(t + 1) * TILE_ROWS * C_DIM);
            wait_tensor<1>();
        } else {
            wait_tensor<0>();
        }

        const float* tile = &buf[wave][t & 1][0];

        #pragma unroll
        for (int r = 0; r < TILE_ROWS; ++r) {
            const float* src = tile + r * C_DIM;

            // ---- per-lane argmax over 32 values (8 x float4, ascending) ----
            float m  = -__builtin_inff();
            int   mi = 0;
            #pragma unroll
            for (int j = 0; j < 8; ++j) {
                const int    bidx = lane * 4 + j * 128;
                const float4 v    = *(const float4*)(src + bidx);
                if (v.x > m) { m = v.x; mi = bidx;     }
                if (v.y > m) { m = v.y; mi = bidx + 1; }
                if (v.z > m) { m = v.z; mi = bidx + 2; }
                if (v.w > m) { m = v.w; mi = bidx + 3; }
            }

            // ---- wave32 butterfly: all lanes get (max, lowest index) ----
            #pragma unroll
            for (int s = 16; s > 0; s >>= 1) {
                const float om = __shfl_xor(m,  s, 32);
                const int   oi = __shfl_xor(mi, s, 32);
                if (om > m || (om == m && oi < mi)) { m = om; mi = oi; }
            }

            const int row = row0 + t * TILE_ROWS + r;
            if (lane == 0) out_idx[row] = (float)mi;

            // ---- gather basis[mi]: 225 float4; nontemporal stores ----
            const v4f* bsrc = (const v4f*)(basis + (size_t)mi * D_DIM);
            v4f*       bdst = (v4f*)(out_q + (size_t)row * D_DIM);
            #pragma unroll
            for (int k = 0; k < 7; ++k) {
                const int e = lane + 32 * k;          // 0..223
                __builtin_nontemporal_store(bsrc[e], &bdst[e]);
            }
            if (lane == 0) __builtin_nontemporal_store(bsrc[224], &bdst[224]);
        }
    }
}

extern "C" void kernel_launch(void* const* d_in, const int* in_sizes, int n_in,
                              void* d_out, int out_size, void* d_ws, size_t ws_size,
                              hipStream_t stream) {
    (void)n_in; (void)out_size; (void)d_ws; (void)ws_size;

    const float* latent = (const float*)d_in[0];   // [16,2048,1024] f32
    const float* basis  = (const float*)d_in[1];   // [1024,900] f32

    const int BK = in_sizes[0] / C_DIM;            // 32768 rows

    float* out_q   = (float*)d_out;                          // [BK,900]
    float* out_idx = (float*)d_out + (size_t)BK * D_DIM;     // [BK]

    dim3 grid(BK / (NWAVES * ROWS_PER_WAVE));      // 512 blocks
    dim3 block(NT);
    hipLaunchKernelGGL(BasisVQ_tdm_rows, grid, block, 0, stream,
                       latent, basis, out_q, out_idx);
}